// TemporalGNN_5102421147849
// MI455X (gfx1250) — compile-verified
//
#include <hip/hip_runtime.h>

typedef __attribute__((ext_vector_type(2))) float v2f;
typedef __attribute__((ext_vector_type(8))) float v8f;

#define DFEAT 128
#define BN_EPS 1e-5f

// ---------------------------------------------------------------------------
// Zero-fill (workspace must be re-initialized every call: graph replays)
// ---------------------------------------------------------------------------
__global__ void fill_zero_kernel(float* __restrict__ p, long n) {
  long i = (long)blockIdx.x * blockDim.x + threadIdx.x;
  if (i < n) p[i] = 0.0f;
}

// ---------------------------------------------------------------------------
// Degree counts: cnt[dst[e]] += 1   (one thread per edge)
// ---------------------------------------------------------------------------
__global__ void count_edges_kernel(const long long* __restrict__ dst,
                                   float* __restrict__ cnt, int E) {
  int e = blockIdx.x * blockDim.x + threadIdx.x;
  if (e < E) atomicAdd(&cnt[(int)dst[e]], 1.0f);
}

// ---------------------------------------------------------------------------
// Edge scatter: agg[dst[e], f] += feat[src[e], f]
// 128 threads per edge (one per feature): coalesced gather, L2 atomics.
// ---------------------------------------------------------------------------
__global__ void scatter_feats_kernel(const float* __restrict__ feat,
                                     const long long* __restrict__ src,
                                     const long long* __restrict__ dst,
                                     float* __restrict__ agg, int E) {
  long tid = (long)blockIdx.x * blockDim.x + threadIdx.x;
  int e = (int)(tid >> 7);
  int f = (int)(tid & (DFEAT - 1));
  if (e >= E) return;
  int s = (int)src[e];
  int d = (int)dst[e];
  atomicAdd(&agg[(long)d * DFEAT + f], feat[(long)s * DFEAT + f]);
}

// ---------------------------------------------------------------------------
// Fused SAGE GEMM:  out = (agg/max(cnt,1)) @ w_l^T + bias + xin @ w_r^T
// One wave32 per 16x16 output tile; V_WMMA_F32_16X16X4_F32, K-loop of 32
// steps per GEMM, both GEMMs fused into one f32 accumulator.
//
// ISA layouts (lane = 16*hi + r, hi=lane>>4, r=lane&15):
//   A (16x4):  vgpr j holds A[r][2*hi + j]
//   B (4x16):  vgpr j holds B[2*hi + j][r]  (B[k][n] = W[o0+n][k0+k])
//   C/D:       vgpr v holds D[v + 8*hi][r]
// ---------------------------------------------------------------------------
__global__ __launch_bounds__(32) void sage_gemm_wmma_kernel(
    const float* __restrict__ agg,   // [n,128] neighbor sums
    const float* __restrict__ cnt,   // [n] degrees
    const float* __restrict__ xin,   // [n,128] root features
    const float* __restrict__ w_l,   // [128,128] row-major [out][in]
    const float* __restrict__ w_r,   // [128,128]
    const float* __restrict__ bias,  // [128]
    float* __restrict__ out,         // [n,128]
    int nrows) {
  const int lane = threadIdx.x;
  const int r    = lane & 15;
  const int hi   = lane >> 4;
  const int m0   = blockIdx.x * 16;
  const int o0   = blockIdx.y * 16;

  int arow = m0 + r;
  if (arow > nrows - 1) arow = nrows - 1;   // clamp (exact tiling for n=50000)
  const float scale = 1.0f / fmaxf(cnt[arow], 1.0f);

  const float* pAgg = agg + (long)arow * DFEAT + 2 * hi;
  const float* pX   = xin + (long)arow * DFEAT + 2 * hi;
  const float* pBL  = w_l + (long)(o0 + r) * DFEAT + 2 * hi;
  const float* pBR  = w_r + (long)(o0 + r) * DFEAT + 2 * hi;

  v8f c = {0.f, 0.f, 0.f, 0.f, 0.f, 0.f, 0.f, 0.f};

#pragma unroll 4
  for (int k0 = 0; k0 < DFEAT; k0 += 4) {
    // mean-aggregation path: A = mean tile, B = w_l^T tile
    v2f a0 = *(const v2f*)(pAgg + k0);
    a0 = a0 * scale;
    v2f b0 = *(const v2f*)(pBL + k0);
    c = __builtin_amdgcn_wmma_f32_16x16x4_f32(false, a0, false, b0,
                                              (short)0, c, false, false);
    // root path: A = x tile, B = w_r^T tile
    v2f a1 = *(const v2f*)(pX + k0);
    v2f b1 = *(const v2f*)(pBR + k0);
    c = __builtin_amdgcn_wmma_f32_16x16x4_f32(false, a1, false, b1,
                                              (short)0, c, false, false);
  }

  // Epilogue: add bias, store 16x16 tile. Hoist the bounds test to one
  // uniform branch so the common (full-tile) path is 8 unconditional,
  // coalesced stores with no EXEC-mask manipulation.
  const float bv = bias[o0 + r];
  float* pOut = out + (long)(m0 + 8 * hi) * DFEAT + o0 + r;
  if (m0 + 16 <= nrows) {
#pragma unroll
    for (int v = 0; v < 8; ++v) {
      pOut[(long)v * DFEAT] = c[v] + bv;
    }
  } else {
#pragma unroll
    for (int v = 0; v < 8; ++v) {
      const int row = m0 + v + 8 * hi;
      if (row < nrows) out[(long)row * DFEAT + o0 + r] = c[v] + bv;
    }
  }
}

// ---------------------------------------------------------------------------
// Column stats for BatchNorm: one block per feature column.
// ---------------------------------------------------------------------------
__global__ void col_stats_kernel(const float* __restrict__ h, int nrows,
                                 float* __restrict__ mu,
                                 float* __restrict__ rstd) {
  __shared__ float ssum[256];
  __shared__ float ssq[256];
  const int f = blockIdx.x;
  float sum = 0.0f, sq = 0.0f;
  for (int r = threadIdx.x; r < nrows; r += blockDim.x) {
    const float v = h[(long)r * DFEAT + f];
    sum += v;
    sq += v * v;
  }
  ssum[threadIdx.x] = sum;
  ssq[threadIdx.x] = sq;
  __syncthreads();
  for (int s = 128; s > 0; s >>= 1) {
    if (threadIdx.x < s) {
      ssum[threadIdx.x] += ssum[threadIdx.x + s];
      ssq[threadIdx.x] += ssq[threadIdx.x + s];
    }
    __syncthreads();
  }
  if (threadIdx.x == 0) {
    const float m   = ssum[0] / (float)nrows;
    const float var = ssq[0] / (float)nrows - m * m;
    mu[f]   = m;
    rstd[f] = rsqrtf(var + BN_EPS);
  }
}

// ---------------------------------------------------------------------------
// BatchNorm affine + ReLU, in place.
// ---------------------------------------------------------------------------
__global__ void bn_relu_kernel(float* __restrict__ h,
                               const float* __restrict__ mu,
                               const float* __restrict__ rstd,
                               const float* __restrict__ gamma,
                               const float* __restrict__ beta, long n) {
  long i = (long)blockIdx.x * blockDim.x + threadIdx.x;
  if (i < n) {
    const int f = (int)(i & (DFEAT - 1));
    const float v = (h[i] - mu[f]) * rstd[f] * gamma[f] + beta[f];
    h[i] = v > 0.0f ? v : 0.0f;
  }
}

// ---------------------------------------------------------------------------
// Host-side orchestration
// ---------------------------------------------------------------------------
extern "C" void kernel_launch(void* const* d_in, const int* in_sizes, int n_in,
                              void* d_out, int out_size, void* d_ws,
                              size_t ws_size, hipStream_t stream) {
  const float*     x     = (const float*)d_in[0];
  const long long* ei    = (const long long*)d_in[1];  // int64 [2, E]
  const float*     w1_l  = (const float*)d_in[2];
  const float*     b1_l  = (const float*)d_in[3];
  const float*     w1_r  = (const float*)d_in[4];
  const float*     w2_l  = (const float*)d_in[5];
  const float*     b2_l  = (const float*)d_in[6];
  const float*     w2_r  = (const float*)d_in[7];
  const float*     gamma = (const float*)d_in[8];
  const float*     beta  = (const float*)d_in[9];

  const int n = in_sizes[0] / DFEAT;  // 50000
  const int E = in_sizes[1] / 2;      // 800000
  const long long* src = ei;
  const long long* dst = ei + E;

  // Workspace layout (floats): agg[n*128] | cnt[n] | h[n*128] | mu[128] | rstd[128]
  float* agg  = (float*)d_ws;
  float* cnt  = agg + (size_t)n * DFEAT;
  float* h    = cnt + n;
  float* mu   = h + (size_t)n * DFEAT;
  float* rstd = mu + DFEAT;

  float* out = (float*)d_out;

  const long hn = (long)n * DFEAT;        // 6.4M floats
  const long st = (long)E * DFEAT;        // scatter threads

  // Layer 1: zero agg+cnt (contiguous), count degrees, scatter x
  fill_zero_kernel<<<(int)((hn + n + 255) / 256), 256, 0, stream>>>(agg, hn + n);
  count_edges_kernel<<<(E + 255) / 256, 256, 0, stream>>>(dst, cnt, E);
  scatter_feats_kernel<<<(int)((st + 255) / 256), 256, 0, stream>>>(x, src, dst, agg, E);

  dim3 ggrid((n + 15) / 16, DFEAT / 16);  // 3125 x 8 waves
  sage_gemm_wmma_kernel<<<ggrid, 32, 0, stream>>>(agg, cnt, x, w1_l, w1_r, b1_l, h, n);

  // BatchNorm (training stats) + ReLU
  col_stats_kernel<<<DFEAT, 256, 0, stream>>>(h, n, mu, rstd);
  bn_relu_kernel<<<(int)((hn + 255) / 256), 256, 0, stream>>>(h, mu, rstd, gamma, beta, hn);

  // Layer 2: re-zero agg, scatter h, final GEMM into d_out
  fill_zero_kernel<<<(int)((hn + 255) / 256), 256, 0, stream>>>(agg, hn);
  scatter_feats_kernel<<<(int)((st + 255) / 256), 256, 0, stream>>>(h, src, dst, agg, E);
  sage_gemm_wmma_kernel<<<ggrid, 32, 0, stream>>>(agg, cnt, h, w2_l, w2_r, b2_l, out, n);
}